// Hitsbe_40510131536188
// MI455X (gfx1250) — compile-verified
//
#include <hip/hip_runtime.h>
#include <math.h>

typedef __attribute__((ext_vector_type(2))) float v2f;
typedef __attribute__((ext_vector_type(8))) float v8f;

#define B_   32
#define TS_  8192
#define S_   256
#define DSEG_ 32
#define DM_  256
#define W_   512

__device__ __forceinline__ float rsum(const float* p, int lo, int n) {
    float s = 0.f;
    for (int i = 0; i < n; ++i) s += p[lo + i];
    return s;
}

__global__ __launch_bounds__(512) void hitsbe_fused(
    const float* __restrict__ x,          // [32][8192]
    const float* __restrict__ vocab,      // [256][512][32]
    const float* __restrict__ word_emb,   // [131072][256]
    const float* __restrict__ haar_emb,   // [7][256]
    float* __restrict__ out)              // [32*256*256] out, then [32*256] mask
{
    __shared__ float bs64[128];       // 64-wide block sums of x[b]
    __shared__ float hA[16 * 8];      // A-tile: 16 rows x 8 K (K=7 padded to 8)
    __shared__ int   widx[16];        // word-emb row per s-row
    __shared__ float tile[16 * 264];  // haar_embed D-tile, padded pitch

    const int b    = blockIdx.y;
    const int st   = blockIdx.x;
    const int s0   = st * 16;
    const int tid  = threadIdx.x;
    const int wave = tid >> 5;
    const int lane = tid & 31;

    // ---------- Phase A: block sums of 64 (Haar building blocks) ----------
    if (tid < 128) {
        const float4* px = (const float4*)(x + (size_t)b * TS_ + tid * 64);
        float s = 0.f;
        #pragma unroll
        for (int i = 0; i < 16; ++i) {
            float4 v = px[i];
            s += v.x + v.y + v.z + v.w;
        }
        bs64[tid] = s;
    }

    // ---------- Phase B: one wave per segment s = s0+wave ----------
    {
        const int s = s0 + wave;
        float seg = x[(size_t)b * TS_ + s * DSEG_ + lane];
        float mn = seg, mx = seg, sum = seg;
        #pragma unroll
        for (int off = 16; off; off >>= 1) {
            mn  = fminf(mn, __shfl_xor(mn, off, 32));
            mx  = fmaxf(mx, __shfl_xor(mx, off, 32));
            sum += __shfl_xor(sum, off, 32);
        }
        const float mean = sum * (1.0f / 32.0f);
        // 4-lane chunk sums -> 8 chunk means vs segment mean -> 8-bit bucket
        float cs = seg + __shfl_xor(seg, 1, 32);
        cs += __shfl_xor(cs, 2, 32);
        const int bit = (cs * 0.25f) > mean;
        const unsigned bmask = (unsigned)__ballot(bit);
        int bucket = 0;
        #pragma unroll
        for (int i = 0; i < 8; ++i) bucket |= ((bmask >> (4 * i)) & 1u) << i;

        const float sn = (seg - mn) / (mx - mn + 1e-8f);
        float snv[32];
        #pragma unroll
        for (int d = 0; d < 32; ++d) snv[d] = __shfl(sn, d, 32);

        // L1 scan: 512 candidates, 16 per lane (c = j*32 + lane)
        const float* vrow = vocab + (size_t)bucket * (W_ * DSEG_);
        float bd = 3.4e38f;
        int   bc = 0x7fffffff;
        for (int j = 0; j < 16; ++j) {
            const int c = j * 32 + lane;
            const float4* cp = (const float4*)(vrow + c * DSEG_);
            float dist = 0.f;
            #pragma unroll
            for (int q = 0; q < 8; ++q) {
                float4 v = cp[q];
                dist += fabsf(v.x - snv[4 * q + 0]) + fabsf(v.y - snv[4 * q + 1])
                      + fabsf(v.z - snv[4 * q + 2]) + fabsf(v.w - snv[4 * q + 3]);
            }
            if (dist < bd) { bd = dist; bc = c; }
        }
        #pragma unroll
        for (int off = 16; off; off >>= 1) {
            float od = __shfl_xor(bd, off, 32);
            int   oc = __shfl_xor(bc, off, 32);
            if (od < bd || (od == bd && oc < bc)) { bd = od; bc = oc; }
        }
        if (lane == 0) widx[wave] = bucket * W_ + bc;
    }
    __syncthreads();

    // ---------- Phase C: Haar coefficient A-tile (16 x 8) ----------
    // taken[k]: k=0 a13, k=1..6 details at levels 13..8 (scaled block sums)
    if (tid < 128) {
        const int m = tid >> 3, k = tid & 7;
        const int s = s0 + m;
        float v = 0.f;
        switch (k) {
        case 0: v = 0.011048543456039806f * rsum(bs64, 0, 128); break;                       // a13: 2^-6.5
        case 1: v = 0.011048543456039806f * (rsum(bs64, 0, 64) - rsum(bs64, 64, 64)); break; // d13
        case 2: { int i = s >> 7; v = 0.015625f            * (rsum(bs64, 64 * i, 32) - rsum(bs64, 64 * i + 32, 32)); } break; // d12: 2^-6
        case 3: { int i = s >> 6; v = 0.02209708691207961f * (rsum(bs64, 32 * i, 16) - rsum(bs64, 32 * i + 16, 16)); } break; // d11: 2^-5.5
        case 4: { int i = s >> 5; v = 0.03125f             * (rsum(bs64, 16 * i,  8) - rsum(bs64, 16 * i +  8,  8)); } break; // d10: 2^-5
        case 5: { int i = s >> 4; v = 0.04419417382415922f * (rsum(bs64,  8 * i,  4) - rsum(bs64,  8 * i +  4,  4)); } break; // d9 : 2^-4.5
        case 6: { int i = s >> 3; v = 0.0625f              * (rsum(bs64,  4 * i,  2) - rsum(bs64,  4 * i +  2,  2)); } break; // d8 : 2^-4
        default: v = 0.f; break; // K pad
        }
        hA[m * 8 + k] = v;
    }
    __syncthreads();

    // ---------- Phase D: haar_embed tile via V_WMMA_F32_16X16X4_F32 x2 (K=8) ----------
    {
        const int h     = lane >> 4;   // lane half
        const int lm    = lane & 15;   // M for A, N for B
        const int nbase = wave * 16;
        const int col   = nbase + lm;

        v2f a0, a1, b0, b1;
        a0.x = hA[lm * 8 + 2 * h];     a0.y = hA[lm * 8 + 2 * h + 1];
        a1.x = hA[lm * 8 + 4 + 2 * h]; a1.y = hA[lm * 8 + 5 + 2 * h];
        b0.x = haar_emb[(2 * h) * DM_ + col];
        b0.y = haar_emb[(2 * h + 1) * DM_ + col];
        b1.x = haar_emb[(4 + 2 * h) * DM_ + col];
        b1.y = h ? 0.f : haar_emb[5 * DM_ + col];   // K=7 is pad -> 0

        v8f c = {0.f, 0.f, 0.f, 0.f, 0.f, 0.f, 0.f, 0.f};
        c = __builtin_amdgcn_wmma_f32_16x16x4_f32(false, a0, false, b0, (short)0, c, false, false);
        c = __builtin_amdgcn_wmma_f32_16x16x4_f32(false, a1, false, b1, (short)0, c, false, false);

        #pragma unroll
        for (int r = 0; r < 8; ++r)
            tile[(r + 8 * h) * 264 + col] = c[r];
    }
    __syncthreads();

    // ---------- Phase E: LN(haar) + word_emb + pos_emb, LN, store ----------
    {
        const int row = tid >> 5;    // one wave per s-row
        const int t   = tid & 31;
        const int s   = s0 + row;

        float h8[8];
        float sum = 0.f;
        #pragma unroll
        for (int j = 0; j < 8; ++j) { h8[j] = tile[row * 264 + t + 32 * j]; sum += h8[j]; }
        #pragma unroll
        for (int off = 16; off; off >>= 1) sum += __shfl_xor(sum, off, 32);
        const float m1 = sum * (1.0f / 256.0f);
        float var = 0.f;
        #pragma unroll
        for (int j = 0; j < 8; ++j) { float d = h8[j] - m1; var += d * d; }
        #pragma unroll
        for (int off = 16; off; off >>= 1) var += __shfl_xor(var, off, 32);
        const float inv1 = rsqrtf(var * (1.0f / 256.0f) + 1e-5f);

        const float* we = word_emb + (size_t)widx[row] * DM_;
        float v8[8];
        float sum2 = 0.f;
        #pragma unroll
        for (int j = 0; j < 8; ++j) {
            const int col = t + 32 * j;
            const float hn = (h8[j] - m1) * inv1;
            const int   c2 = col & ~1;
            const float ang = (float)s * expf((float)c2 * (-9.210340371976184f / 256.0f));
            const float pe  = (col & 1) ? cosf(ang) : sinf(ang);
            const float val = hn + we[col] + pe;
            v8[j] = val;
            sum2 += val;
        }
        #pragma unroll
        for (int off = 16; off; off >>= 1) sum2 += __shfl_xor(sum2, off, 32);
        const float m2 = sum2 * (1.0f / 256.0f);
        float var2 = 0.f;
        #pragma unroll
        for (int j = 0; j < 8; ++j) { float d = v8[j] - m2; var2 += d * d; }
        #pragma unroll
        for (int off = 16; off; off >>= 1) var2 += __shfl_xor(var2, off, 32);
        const float inv2 = rsqrtf(var2 * (1.0f / 256.0f) + 1e-5f);

        float* orow = out + ((size_t)b * S_ + s) * DM_;
        #pragma unroll
        for (int j = 0; j < 8; ++j)
            orow[t + 32 * j] = (v8[j] - m2) * inv2;
    }

    // ---------- att_mask (all ones) ----------
    if (tid < 16)
        out[(size_t)B_ * S_ * DM_ + b * S_ + s0 + tid] = 1.0f;
}

extern "C" void kernel_launch(void* const* d_in, const int* in_sizes, int n_in,
                              void* d_out, int out_size, void* d_ws, size_t ws_size,
                              hipStream_t stream) {
    (void)in_sizes; (void)n_in; (void)d_ws; (void)ws_size; (void)out_size;
    const float* x        = (const float*)d_in[0];
    const float* vocab    = (const float*)d_in[1];
    const float* word_emb = (const float*)d_in[2];
    const float* haar_emb = (const float*)d_in[3];
    float* out = (float*)d_out;

    dim3 grid(S_ / 16, B_);   // (16 s-tiles, 32 batches)
    hitsbe_fused<<<grid, 512, 0, stream>>>(x, vocab, word_emb, haar_emb, out);
}